// AncProbsLayer_75831942578745
// MI455X (gfx1250) — compile-verified
//
#include <hip/hip_runtime.h>
#include <hip/hip_bf16.h>

// Problem constants (from reference setup_inputs)
#define M_  2
#define B_  256
#define L_  1024
#define S_  20
#define K_  4
#define NR_ 100

typedef __attribute__((ext_vector_type(2))) float v2f;
typedef __attribute__((ext_vector_type(8))) float v8f;

__device__ __forceinline__ float softplusf(float x) {
    // numerically stable log(1 + exp(x))
    return fmaxf(x, 0.0f) + log1pf(expf(-fabsf(x)));
}

// ---------------------------------------------------------------------------
// Kernel 1: build normalized rate matrices Q (m,k,20,20)
// one wave (32 threads) per (m,k)
// ---------------------------------------------------------------------------
__global__ void anc_qmat_kernel(const float* __restrict__ exch,   // (m,k,20,20)
                                const float* __restrict__ eq,     // (m,k,20)
                                float* __restrict__ Qout)         // ws (m,k,400)
{
    __shared__ float R[S_ * S_];
    __shared__ float p[S_];
    __shared__ float rowdiag[S_];
    __shared__ float mue_s;

    const int mk   = blockIdx.x;               // 0..7
    const int lane = threadIdx.x;              // 0..31
    const float* Kb = exch + (size_t)mk * (S_ * S_);
    const float* eb = eq   + (size_t)mk * S_;

    // R = softplus(0.5*(K + K^T)), zero diagonal
    for (int idx = lane; idx < S_ * S_; idx += 32) {
        int i = idx / S_, j = idx % S_;
        float v = 0.5f * (Kb[i * S_ + j] + Kb[j * S_ + i]);
        R[idx] = (i == j) ? 0.0f : softplusf(v);
    }
    __syncthreads();

    // p = softmax(eq) (each of lanes 0..19 computes redundantly; tiny)
    if (lane < S_) {
        float mx = -3.0e38f;
        for (int j = 0; j < S_; ++j) mx = fmaxf(mx, eb[j]);
        float sum = 0.0f;
        for (int j = 0; j < S_; ++j) sum += expf(eb[j] - mx);
        p[lane] = expf(eb[lane] - mx) / sum;
    }
    __syncthreads();

    // rowdiag[i] = sum_j R[i][j] * p[j]
    if (lane < S_) {
        float d = 0.0f;
        for (int j = 0; j < S_; ++j) d += R[lane * S_ + j] * p[j];
        rowdiag[lane] = d;
    }
    __syncthreads();

    if (lane == 0) {
        float mue = 0.0f;
        for (int i = 0; i < S_; ++i) mue += p[i] * rowdiag[i];
        mue_s = fmaxf(mue, 1e-16f);
    }
    __syncthreads();

    const float inv = 1.0f / mue_s;
    float* Qb = Qout + (size_t)mk * (S_ * S_);
    for (int idx = lane; idx < S_ * S_; idx += 32) {
        int i = idx / S_, j = idx % S_;
        float q = (i == j) ? -rowdiag[i] : (R[idx] * p[j]);
        Qb[idx] = q * inv;
    }
}

// ---------------------------------------------------------------------------
// Kernel 2: P = expm(mut * Q) via scaling(2^-6) + Taylor-14 + 6 squarings
// one block (128 threads) per (m,b,k) matrix; matrices live in LDS
// ---------------------------------------------------------------------------
__global__ void anc_expm_kernel(const float* __restrict__ Q,      // ws (m,k,400)
                                const float* __restrict__ tauk,   // (m,nr)
                                const float* __restrict__ pmrk,   // (m,k)
                                const int*   __restrict__ ridx,   // (m,b)
                                float* __restrict__ Pout)         // ws (m,b,k,400)
{
    __shared__ float A[400];
    __shared__ float O0[400];
    __shared__ float O1[400];
    __shared__ float T0[400];
    __shared__ float T1[400];

    const int blk = blockIdx.x;                // 0..2047
    const int mi  = blk / (B_ * K_);
    const int rem = blk % (B_ * K_);
    const int bi  = rem / K_;
    const int ki  = rem % K_;
    const int tid = threadIdx.x;               // 0..127

    const int r_idx = ridx[mi * B_ + bi];
    const float tau = softplusf(tauk[mi * NR_ + r_idx]);
    const float pm  = softplusf(pmrk[mi * K_ + ki]);
    const float mut = tau * pm;

    const float* Qb = Q + (size_t)(mi * K_ + ki) * 400;
    for (int e = tid; e < 400; e += 128) {
        A[e] = Qb[e] * mut * (1.0f / 64.0f);    // 2^-6 scaling
        int i = e / S_, j = e % S_;
        float id = (i == j) ? 1.0f : 0.0f;
        O0[e] = id;                              // out accumulator = I
        T0[e] = id;                              // current term   = I
    }
    __syncthreads();

    // Taylor series: term = term @ A / i ; out += term
    float* ts = T0;
    float* td = T1;
    for (int it = 1; it <= 14; ++it) {
        float rinv = 1.0f / (float)it;
        for (int e = tid; e < 400; e += 128) {
            int r = e / S_, c = e % S_;
            float acc = 0.0f;
            #pragma unroll
            for (int t = 0; t < S_; ++t) acc += ts[r * S_ + t] * A[t * S_ + c];
            td[e] = acc * rinv;
        }
        __syncthreads();
        for (int e = tid; e < 400; e += 128) O0[e] += td[e];
        __syncthreads();
        float* tmp = ts; ts = td; td = tmp;
    }

    // 6 squarings (ping-pong O0/O1; even count -> result ends in s0 == O0)
    float* s0 = O0;
    float* s1 = O1;
    for (int sq = 0; sq < 6; ++sq) {
        for (int e = tid; e < 400; e += 128) {
            int r = e / S_, c = e % S_;
            float acc = 0.0f;
            #pragma unroll
            for (int t = 0; t < S_; ++t) acc += s0[r * S_ + t] * s0[t * S_ + c];
            s1[e] = acc;
        }
        __syncthreads();
        float* tmp = s0; s0 = s1; s1 = tmp;
    }

    float* Pb = Pout + (size_t)blk * 400;
    for (int e = tid; e < 400; e += 128) Pb[e] = s0[e];
}

// ---------------------------------------------------------------------------
// Kernel 3: anc[l, k*20+s] = sum_z in[l,z] * P[k,z,s]      (WMMA f32 16x16x4)
// One block (256 thr = 8 waves) per (m,b). P panel staged once via
// global_load_async_to_lds_b128 (ASYNCcnt); all 25 B fragments hoisted to
// registers; each wave walks 8 M-tiles of 16 rows.
// A-frag layout (ISA 16x4 f32): lanes 0-15 -> K={0,1}, lanes 16-31 -> K={2,3}.
// B-frag layout (mirrors C/D half-lane split): VGPR v, half h -> K = v + 2h.
// D layout (ISA 16x16 f32): VGPR v, half h -> row = v + 8h, col = lane%16.
// ---------------------------------------------------------------------------
__global__ void __launch_bounds__(256)
anc_einsum_wmma_kernel(const float* __restrict__ in,    // (m,b,L,20)
                       const float* __restrict__ P,     // ws (m,b,k,20,20)
                       float* __restrict__ out)         // (m,b,L,80)
{
    __shared__ float Bs[K_ * S_ * S_];        // 1600 f32 = 6.4 KB: P[m,b] panel

    const int mb   = blockIdx.x;              // 0..511  (m*b)
    const int tid  = threadIdx.x;             // 0..255
    const int wave = tid >> 5;                // 0..7
    const int lane = tid & 31;
    const int ln16 = lane & 15;
    const int h    = lane >> 4;               // half-wave: 0 or 1

    // ---- stage the per-sequence P panel into LDS via async-to-LDS DMA ----
    const float* Pb = P + (size_t)mb * (K_ * S_ * S_);
    for (int i = tid; i < 400; i += 256) {    // 400 x b128 = 6400 bytes
        unsigned            ldsb = (unsigned)(size_t)Bs + (unsigned)(i * 16);
        unsigned long long  ga   = (unsigned long long)(size_t)Pb +
                                   (unsigned long long)(i * 16);
        asm volatile("global_load_async_to_lds_b128 %0, %1, off"
                     :: "v"(ldsb), "v"(ga) : "memory");
    }
    asm volatile("s_wait_asynccnt 0" ::: "memory");
    __syncthreads();

    // ---- hoist all 25 B fragments (k-chunk x n-tile) into registers ----
    v2f bfr[5][5];
    #pragma unroll
    for (int nt = 0; nt < 5; ++nt) {
        const int n    = nt * 16 + ln16;      // output column 0..79
        const int kcol = n / S_;
        const int scol = n % S_;
        const float* Bcol = Bs + kcol * (S_ * S_) + scol;
        #pragma unroll
        for (int kc = 0; kc < 5; ++kc) {
            const int z0 = kc * 4 + 2 * h;    // this half-wave's K rows
            bfr[nt][kc].x = Bcol[z0 * S_];    // B[z0][n]
            bfr[nt][kc].y = Bcol[z0 * S_ + S_];
        }
    }

    const float* Abase = in + (size_t)mb * (L_ * S_);
    float*       Ob    = out + (size_t)mb * (L_ * (K_ * S_));

    // ---- 8 M-tiles per wave: pure load -> 25 WMMA -> NT-store pipeline ----
    for (int mt = wave; mt < L_ / 16; mt += 8) {
        const int row = mt * 16 + ln16;
        const float* Ab = Abase + (size_t)row * S_;
        if (mt + 8 < L_ / 16) {               // uniform per wave
            __builtin_prefetch(Abase + (size_t)(row + 128) * S_, 0, 0);
        }

        v2f a[5];
        #pragma unroll
        for (int kc = 0; kc < 5; ++kc) {
            const int col = kc * 4 + h * 2;   // contiguous pair -> b64 load
            a[kc].x = Ab[col];
            a[kc].y = Ab[col + 1];
        }

        const int rbase = mt * 16 + 8 * h;
        #pragma unroll
        for (int nt = 0; nt < 5; ++nt) {
            v8f c = {};
            #pragma unroll
            for (int kc = 0; kc < 5; ++kc) {
                c = __builtin_amdgcn_wmma_f32_16x16x4_f32(
                        false, a[kc], false, bfr[nt][kc], (short)0, c,
                        false, false);
            }
            const int n = nt * 16 + ln16;
            #pragma unroll
            for (int v = 0; v < 8; ++v) {
                // write-once 168MB stream: NT hint keeps read side in L2
                __builtin_nontemporal_store(
                    c[v], &Ob[(size_t)(rbase + v) * (K_ * S_) + n]);
            }
        }
    }
}

// ---------------------------------------------------------------------------
extern "C" void kernel_launch(void* const* d_in, const int* in_sizes, int n_in,
                              void* d_out, int out_size, void* d_ws, size_t ws_size,
                              hipStream_t stream) {
    const float* inputs = (const float*)d_in[0];   // (m,b,L,20)
    const float* tauk   = (const float*)d_in[1];   // (m,nr)
    const float* exch   = (const float*)d_in[2];   // (m,k,20,20)
    const float* eq     = (const float*)d_in[3];   // (m,k,20)
    const float* pmrk   = (const float*)d_in[4];   // (m,k)
    const int*   ridx   = (const int*)d_in[5];     // (m,b) int32

    float* Qws = (float*)d_ws;                         // 3200 floats
    float* Pws = Qws + (size_t)M_ * K_ * S_ * S_;      // m*b*k*400 floats
    float* out = (float*)d_out;                        // (m,b,L,80)

    // 1) rate matrices
    anc_qmat_kernel<<<M_ * K_, 32, 0, stream>>>(exch, eq, Qws);

    // 2) matrix exponentials
    anc_expm_kernel<<<M_ * B_ * K_, 128, 0, stream>>>(Qws, tauk, pmrk, ridx, Pws);

    // 3) streaming einsum via WMMA
    anc_einsum_wmma_kernel<<<M_ * B_, 256, 0, stream>>>(inputs, Pws, out);
}